// MyGNN_57896159150675
// MI455X (gfx1250) — compile-verified
//
#include <hip/hip_runtime.h>

typedef float v2f __attribute__((ext_vector_type(2)));
typedef float v4f __attribute__((ext_vector_type(4)));
typedef float v8f __attribute__((ext_vector_type(8)));

static constexpr float NEG_SLOPE = 0.01f;

__device__ __forceinline__ v8f wmma4(v2f a, v2f b, v8f c) {
  // V_WMMA_F32_16X16X4_F32 : D = A(16x4) * B(4x16) + C(16x16)
  return __builtin_amdgcn_wmma_f32_16x16x4_f32(
      /*neg_a=*/false, a, /*neg_b=*/false, b,
      /*c_mod=*/(short)0, c, /*reuse_a=*/false, /*reuse_b=*/false);
}

__global__ __launch_bounds__(256) void gine_zero_kernel(v4f* __restrict__ p, int n4) {
  int i = blockIdx.x * blockDim.x + threadIdx.x;
  const int stride = gridDim.x * blockDim.x;
  const v4f z = (v4f){0.f, 0.f, 0.f, 0.f};
  for (; i < n4; i += stride) p[i] = z;
}

// One wave handles 16 edges: e = relu(x[src] + ea@We + be); atomicAdd into agg[dst].
__global__ __launch_bounds__(256) void gine_edge_kernel(
    const float* __restrict__ x,    // [N,32] current node features
    const float* __restrict__ ea,   // [E,8]
    const float* __restrict__ We,   // [8,32]  (layer slice)
    const float* __restrict__ be,   // [32]
    const int*   __restrict__ src,  // [E]
    const int*   __restrict__ dst,  // [E]
    float*       __restrict__ agg,  // [N,32] accumulator
    int n_edges)
{
  const int wave_id = (int)((blockIdx.x * blockDim.x + threadIdx.x) >> 5);
  const int lane    = threadIdx.x & 31;
  const int ebase   = wave_id * 16;
  if (ebase >= n_edges) return;           // wave-uniform: EXEC stays all-ones

  const int hi = lane >> 4;               // half-wave select
  const int lo = lane & 15;

  // ---- A fragments: lane holds ea[row][4s+2*hi .. +1] for steps s=0,1
  int arow = ebase + lo;
  if (arow >= n_edges) arow = n_edges - 1;   // clamp (tail rows discarded below)
  const float* ap = ea + arow * 8 + 2 * hi;
  const v2f a0 = *(const v2f*)(ap);          // K = 2hi, 2hi+1
  const v2f a1 = *(const v2f*)(ap + 4);      // K = 4+2hi, 4+2hi+1

  // ---- B fragments + C init (bias broadcast)
  v8f c[2];
  v2f b0[2], b1[2];
#pragma unroll
  for (int nt = 0; nt < 2; ++nt) {
    const int col = lo + 16 * nt;
    const int k0 = 2 * hi;
    const int k1 = 4 + 2 * hi;
    b0[nt] = (v2f){We[k0 * 32 + col], We[(k0 + 1) * 32 + col]};
    b1[nt] = (v2f){We[k1 * 32 + col], We[(k1 + 1) * 32 + col]};
    const float bv = be[col];
    c[nt] = (v8f){bv, bv, bv, bv, bv, bv, bv, bv};
  }

#pragma unroll
  for (int nt = 0; nt < 2; ++nt) {
    c[nt] = wmma4(a0, b0[nt], c[nt]);
    c[nt] = wmma4(a1, b1[nt], c[nt]);
  }

  // ---- gather x[src], ReLU, scatter-add into agg[dst]
  // C layout: vgpr i, lanes 0-15 -> row ebase+i, col=lo ; lanes 16-31 -> row ebase+i+8
  if (ebase + 16 <= n_edges) {
    // fast path: full tile, no predication, 32-bit scaled indexing
#pragma unroll
    for (int i = 0; i < 8; ++i) {
      const int erow = ebase + i + 8 * hi;
      const int si = src[erow] * 32 + lo;   // element index (fits 32-bit)
      const int di = dst[erow] * 32 + lo;
      float v0 = fmaxf(x[si]      + c[0][i], 0.0f);
      float v1 = fmaxf(x[si + 16] + c[1][i], 0.0f);
      atomicAdd(&agg[di],      v0);
      atomicAdd(&agg[di + 16], v1);
    }
  } else {
    // tail path: predicate per edge row
#pragma unroll
    for (int i = 0; i < 8; ++i) {
      const int  erow  = ebase + i + 8 * hi;
      const bool valid = (erow < n_edges);
      const int  si = (valid ? src[erow] : 0) * 32 + lo;
      const int  di = (valid ? dst[erow] : 0) * 32 + lo;
      float v0 = fmaxf(x[si]      + c[0][i], 0.0f);
      float v1 = fmaxf(x[si + 16] + c[1][i], 0.0f);
      if (valid) {
        atomicAdd(&agg[di],      v0);
        atomicAdd(&agg[di + 16], v1);
      }
    }
  }
}

// One wave handles 16 nodes: out = leaky_relu((x+agg) @ W + b). out may alias x.
__global__ __launch_bounds__(256) void gine_node_kernel(
    const float* __restrict__ x,    // [N,32]
    const float* __restrict__ agg,  // [N,32]
    const float* __restrict__ W,    // [32,32] (layer slice)
    const float* __restrict__ b,    // [32]
    float*       __restrict__ out,  // [N,32]
    int n_nodes)
{
  const int wave_id = (int)((blockIdx.x * blockDim.x + threadIdx.x) >> 5);
  const int lane    = threadIdx.x & 31;
  const int nbase   = wave_id * 16;
  if (nbase >= n_nodes) return;           // wave-uniform

  const int hi = lane >> 4;
  const int lo = lane & 15;

  int arow = nbase + lo;
  if (arow >= n_nodes) arow = n_nodes - 1;
  const float* xr = x   + arow * 32;
  const float* gr = agg + arow * 32;

  v8f c[2];
#pragma unroll
  for (int nt = 0; nt < 2; ++nt) {
    const float bv = b[lo + 16 * nt];
    c[nt] = (v8f){bv, bv, bv, bv, bv, bv, bv, bv};
  }

#pragma unroll
  for (int s = 0; s < 8; ++s) {           // K = 32 in 8 steps of 4
    const int k = 4 * s + 2 * hi;         // even -> 8B-aligned v2f loads
    const v2f xv = *(const v2f*)(xr + k);
    const v2f gv = *(const v2f*)(gr + k);
    const v2f a  = (v2f){xv.x + gv.x, xv.y + gv.y};
#pragma unroll
    for (int nt = 0; nt < 2; ++nt) {
      const int col = lo + 16 * nt;
      const v2f bb = (v2f){W[k * 32 + col], W[(k + 1) * 32 + col]};
      c[nt] = wmma4(a, bb, c[nt]);
    }
  }

  if (nbase + 16 <= n_nodes) {
#pragma unroll
    for (int i = 0; i < 8; ++i) {
      const int ri = (nbase + i + 8 * hi) * 32 + lo;
#pragma unroll
      for (int nt = 0; nt < 2; ++nt) {
        float v = c[nt][i];
        v = (v > 0.0f) ? v : v * NEG_SLOPE;
        out[ri + 16 * nt] = v;
      }
    }
  } else {
#pragma unroll
    for (int i = 0; i < 8; ++i) {
      const int r = nbase + i + 8 * hi;
      if (r < n_nodes) {
#pragma unroll
        for (int nt = 0; nt < 2; ++nt) {
          float v = c[nt][i];
          v = (v > 0.0f) ? v : v * NEG_SLOPE;
          out[r * 32 + lo + 16 * nt] = v;
        }
      }
    }
  }
}

extern "C" void kernel_launch(void* const* d_in, const int* in_sizes, int n_in,
                              void* d_out, int out_size, void* d_ws, size_t ws_size,
                              hipStream_t stream) {
  const float* x0  = (const float*)d_in[0];  // [N,32]
  const float* ea  = (const float*)d_in[1];  // [E,8]
  const float* W   = (const float*)d_in[2];  // [3,32,32]
  const float* b   = (const float*)d_in[3];  // [3,32]
  const float* We  = (const float*)d_in[4];  // [3,8,32]
  const float* be  = (const float*)d_in[5];  // [3,32]
  const int*  eidx = (const int*)d_in[6];    // [2,E]

  const int n_nodes = in_sizes[0] / 32;
  const int n_edges = in_sizes[1] / 8;
  const int* src = eidx;
  const int* dst = eidx + n_edges;

  float* agg  = (float*)d_ws;                // n_nodes*32 floats (12.8 MB)
  float* xout = (float*)d_out;               // doubles as x-buffer for layers 1..2

  const int feat4 = (n_nodes * 32) / 4;      // N*32 always divisible by 4
  int zero_blocks = (feat4 + 255) / 256;
  if (zero_blocks > 2048) zero_blocks = 2048;

  const int edge_waves  = (n_edges + 15) / 16;
  const int edge_blocks = (edge_waves + 7) / 8;    // 8 waves / 256-thread block
  const int node_waves  = (n_nodes + 15) / 16;
  const int node_blocks = (node_waves + 7) / 8;

  const float* xcur = x0;
  for (int l = 0; l < 3; ++l) {
    gine_zero_kernel<<<zero_blocks, 256, 0, stream>>>((v4f*)agg, feat4);
    gine_edge_kernel<<<edge_blocks, 256, 0, stream>>>(
        xcur, ea, We + (size_t)l * 8 * 32, be + (size_t)l * 32,
        src, dst, agg, n_edges);
    gine_node_kernel<<<node_blocks, 256, 0, stream>>>(
        xcur, agg, W + (size_t)l * 32 * 32, b + (size_t)l * 32,
        xout, n_nodes);
    xcur = xout;
  }
}